// MyGCNConv_36515811950755
// MI455X (gfx1250) — compile-verified
//
#include <hip/hip_runtime.h>
#include <stdint.h>

#define DIN 128
#define DOUT 128

typedef float v2f __attribute__((ext_vector_type(2)));
typedef float v8f __attribute__((ext_vector_type(8)));

// ---------------- utility: zero a u32 region (grid-stride) ----------------
__global__ void k_zero(uint32_t* __restrict__ p, size_t n) {
  size_t i = (size_t)blockIdx.x * blockDim.x + threadIdx.x;
  size_t stride = (size_t)gridDim.x * blockDim.x;
  for (; i < n; i += stride) p[i] = 0u;
}

// ------------- pass 1: dedup edges via bitmap, count distinct deg ---------
__global__ void k_degree(const int* __restrict__ src, const int* __restrict__ dst,
                         int E, int N, uint32_t* bitmap, uint32_t* __restrict__ degCnt) {
  int e = blockIdx.x * blockDim.x + threadIdx.x;
  if (e >= E) return;
  int s = src[e], d = dst[e];
  if (bitmap) {
    uint32_t b = (uint32_t)s * (uint32_t)N + (uint32_t)d;  // N*N = 2^28 fits u32
    uint32_t m = 1u << (b & 31u);
    uint32_t old = atomicOr(&bitmap[b >> 5], m);
    if ((old & m) == 0u) atomicAdd(&degCnt[s], 1u);
  } else {
    atomicAdd(&degCnt[s], 1u);
  }
}

// ------------- deg[i] = distinct + 1 (self loop); dinv = rsqrt ------------
__global__ void k_dinv(const uint32_t* __restrict__ degCnt, float* __restrict__ dinv, int N) {
  int i = blockIdx.x * blockDim.x + threadIdx.x;
  if (i < N) dinv[i] = rsqrtf((float)(degCnt[i] + 1u));
}

// ------ pass 2: scatter s[src] += dinv[dst] * x[dst]  (wave32 per edge) ---
__global__ void k_scatter(const float* __restrict__ x, const int* __restrict__ src,
                          const int* __restrict__ dst, int E, int N,
                          uint32_t* bitmap, const float* __restrict__ dinv,
                          float* __restrict__ s) {
  int t = blockIdx.x * blockDim.x + threadIdx.x;
  int e = t >> 5;      // wave32: one wave per edge
  int lane = t & 31;
  if (e >= E) return;
  int si = src[e], di = dst[e];
  int own = 1;
  if (bitmap) {
    uint32_t f = 0u;
    if (lane == 0) {
      uint32_t b = (uint32_t)si * (uint32_t)N + (uint32_t)di;
      uint32_t m = 1u << (b & 31u);
      uint32_t old = atomicOr(&bitmap[b >> 5], m);
      f = ((old & m) == 0u) ? 1u : 0u;   // this instance owns the distinct edge
    }
    own = (int)__shfl(f, 0, 32);
  }
  if (!own) return;
  float sc = dinv[di];
  const float4* xr = (const float4*)(x + (size_t)di * DIN);
  float4 xv = xr[lane];                       // lane covers 4 of 128 features
  float* sp = s + (size_t)si * DIN + lane * 4;
  atomicAdd(sp + 0, sc * xv.x);
  atomicAdd(sp + 1, sc * xv.y);
  atomicAdd(sp + 2, sc * xv.z);
  atomicAdd(sp + 3, sc * xv.w);
}

// -------- ax[i,k] = dinv[i] * (s[i,k] + dinv[i]*x[i,k])  (in place) -------
__global__ void k_finalize(const float* __restrict__ x, const float* __restrict__ dinv,
                           float* __restrict__ s, int total) {
  int i = blockIdx.x * blockDim.x + threadIdx.x;
  if (i >= total) return;
  int row = i >> 7;                           // /DIN
  float di = dinv[row];
  s[i] = di * (s[i] + di * x[i]);
}

// -------- out = tanh(ax @ W) with native f32 WMMA (16x16x4), async LDS ----
__global__ void __launch_bounds__(256) k_gemm_tanh(const float* __restrict__ A,
                                                   const float* __restrict__ W,
                                                   float* __restrict__ out) {
  __shared__ float sw[DIN * DOUT];            // 64 KB of weights in LDS
  const int tid = threadIdx.x;

  // Stage W into LDS with gfx1250 async global->LDS copies (ASYNCcnt path).
  for (int i = tid; i < (DIN * DOUT) / 4; i += 256) {
    uint32_t lds = (uint32_t)(uintptr_t)(&sw[i * 4]);  // low 32 bits = LDS offset
    uint32_t off = (uint32_t)(i * 16);                 // byte offset into W
    asm volatile("global_load_async_to_lds_b128 %0, %1, %2"
                 :: "v"(lds), "v"(off), "s"(W)
                 : "memory");
  }
  asm volatile("s_wait_asynccnt 0" ::: "memory");
  __syncthreads();

  const int wave = tid >> 5;                  // 8 waves -> 8 column tiles
  const int lane = tid & 31;
  const int rowBase = blockIdx.x * 16;
  const int colBase = wave * 16;
  const int m = lane & 15;                    // M for A-frag, N for B/C-frag
  const int kk = (lane >> 4) * 2;             // lanes 0-15: K=0,1; 16-31: K=2,3

  const float* arow = A + (size_t)(rowBase + m) * DIN;
  const float* bcol = sw + colBase + m;

  v8f c = {};
#pragma unroll
  for (int k0 = 0; k0 < DIN; k0 += 4) {
    v2f a, b;
    a.x = arow[k0 + kk];
    a.y = arow[k0 + kk + 1];
    b.x = bcol[(k0 + kk) * DOUT];
    b.y = bcol[(k0 + kk + 1) * DOUT];
    c = __builtin_amdgcn_wmma_f32_16x16x4_f32(false, a, false, b,
                                              (short)0, c, false, false);
  }

  const int half = lane >> 4;                 // C/D layout: vgpr r -> M = r + 8*half
#pragma unroll
  for (int r = 0; r < 8; ++r) {
    int row = rowBase + half * 8 + r;
    out[(size_t)row * DOUT + colBase + m] = tanhf(c[r]);
  }
}

extern "C" void kernel_launch(void* const* d_in, const int* in_sizes, int n_in,
                              void* d_out, int out_size, void* d_ws, size_t ws_size,
                              hipStream_t stream) {
  const float* x  = (const float*)d_in[0];     // [N, 128] f32
  const int*   ei = (const int*)d_in[1];       // [2, E]   (row 0 = src, row 1 = dst)
  const float* w  = (const float*)d_in[2];     // [128, 128] f32
  float* out = (float*)d_out;                  // [N, 128] f32

  const int N = in_sizes[0] / DIN;             // 16384
  const int E = in_sizes[1] / 2;               // 262144

  // workspace layout: s (N*128 f32) | degCnt (N u32) | dinv (N f32) | bitmap
  uint8_t* ws = (uint8_t*)d_ws;
  float* s = (float*)ws;
  size_t sB = (size_t)N * DIN * sizeof(float);
  uint32_t* degCnt = (uint32_t*)(ws + sB);
  size_t dB = (size_t)N * sizeof(uint32_t);
  float* dinv = (float*)(ws + sB + dB);
  size_t need = sB + 2 * dB;
  size_t bmWords = ((size_t)N * (size_t)N + 31) / 32;   // 32 MB for N=16384
  uint32_t* bitmap = nullptr;
  if (ws_size >= need + bmWords * sizeof(uint32_t))
    bitmap = (uint32_t*)(ws + need);

  const int* srcs = ei;
  const int* dsts = ei + E;

  // zero s + degCnt (contiguous), then bitmap
  size_t zw = (size_t)N * DIN + (size_t)N;
  k_zero<<<1024, 256, 0, stream>>>((uint32_t*)ws, zw);
  if (bitmap) k_zero<<<4096, 256, 0, stream>>>(bitmap, bmWords);

  k_degree<<<(E + 255) / 256, 256, 0, stream>>>(srcs, dsts, E, N, bitmap, degCnt);
  k_dinv<<<(N + 255) / 256, 256, 0, stream>>>(degCnt, dinv, N);

  if (bitmap) k_zero<<<4096, 256, 0, stream>>>(bitmap, bmWords);  // reuse as owner map
  long long totThreads = (long long)E * 32;
  k_scatter<<<(int)((totThreads + 255) / 256), 256, 0, stream>>>(
      x, srcs, dsts, E, N, bitmap, dinv, s);

  int tot = N * DIN;
  k_finalize<<<(tot + 255) / 256, 256, 0, stream>>>(x, dinv, s, tot);

  k_gemm_tanh<<<N / 16, 256, 0, stream>>>(s, w, out);
}